// GBSTEmbedding_54339926229157
// MI455X (gfx1250) — compile-verified
//
#include <hip/hip_runtime.h>
#include <hip/hip_bf16.h>

typedef float v2f __attribute__((ext_vector_type(2)));
typedef float v8f __attribute__((ext_vector_type(8)));

#define VOCAB 256
#define DCH   512
#define KC    5
#define TL    48      // tile length: multiple of lcm(1,2,3,4)=12 and DS=2
#define LLEN  8192

// -----------------------------------------------------------------------------
// Kernel 1: T[k][v][o] = sum_i emb[v][i] * conv_w[o][i][k]
// One wave computes one 16x16 tile of T for one k, via V_WMMA_F32_16X16X4_F32.
// conv_w layout: [o][i][k] (stride K over k).
// -----------------------------------------------------------------------------
__global__ __launch_bounds__(32) void gbst_table_wmma(
    const float* __restrict__ emb,     // [VOCAB][DCH]
    const float* __restrict__ conv_w,  // [DCH][DCH][KC]
    float* __restrict__ T)             // [KC][VOCAB][DCH]
{
  const int k    = blockIdx.z;
  const int v0   = blockIdx.y * 16;
  const int o0   = blockIdx.x * 16;
  const int lane = threadIdx.x;   // 0..31
  const int half = lane >> 4;     // 0 | 1
  const int sub  = lane & 15;
  const int kh   = half * 2;      // K-offset handled by this lane half

  v8f c = {};
  const float* arow = emb + (size_t)(v0 + sub) * DCH;                 // A row M=sub
  const float* bcol = conv_w + (size_t)(o0 + sub) * (DCH * KC) + k;   // B col N=sub

  for (int i0 = 0; i0 < DCH; i0 += 4) {
    v2f a, b;
    // A 16x4 f32: lanes 0-15 hold K={0,1}, lanes 16-31 hold K={2,3}
    a[0] = arow[i0 + kh];
    a[1] = arow[i0 + kh + 1];
    // B 4x16 f32: lanes 0-15 hold K={0,1}, lanes 16-31 hold K={2,3}
    b[0] = bcol[(size_t)(i0 + kh) * KC];
    b[1] = bcol[(size_t)(i0 + kh + 1) * KC];
    c = __builtin_amdgcn_wmma_f32_16x16x4_f32(false, a, false, b,
                                              (short)0, c, false, false);
  }
  // C/D layout: VGPR r -> row (half*8 + r), col = sub
#pragma unroll
  for (int r = 0; r < 8; ++r) {
    T[((size_t)k * VOCAB + (v0 + half * 8 + r)) * DCH + (o0 + sub)] = c[r];
  }
}

// -----------------------------------------------------------------------------
// Kernel 2: fused conv-gather + GBST pooling/softmax/mix + DS=2 mean pool.
// One block = one batch row x 48-position tile. Conv output tile kept in LDS.
// -----------------------------------------------------------------------------
__global__ __launch_bounds__(256) void gbst_main(
    const int*   __restrict__ ids,     // [B][L]
    const float* __restrict__ T,       // [KC][VOCAB][DCH]
    const float* __restrict__ conv_b,  // [DCH]
    const float* __restrict__ sw,      // [DCH] score_w
    float*       __restrict__ out,     // [B][L/2][DCH]
    int L)
{
  __shared__ float xt[TL][DCH];    // conv output tile  (96 KB)
  __shared__ float q[TL];          // per-position score dot
  __shared__ float wc[TL][4];      // softmax weight / m
  __shared__ int   sid[TL + 4];    // ids with conv halo of 2

  const int b   = blockIdx.y;
  const int t0  = blockIdx.x * TL;
  const int tid = threadIdx.x;

  // ---- load ids (halo 2 each side); OOB -> 0 (T row 0 is all-zero) ----
  if (tid < TL + 4) {
    int g = t0 + tid - 2;
    sid[tid] = (g >= 0 && g < L) ? ids[(size_t)b * L + g] : 0;
  }
  __syncthreads();

  const int oc = (tid & 127) * 4;  // channel chunk (float4)
  const int pr = tid >> 7;         // 0 | 1 position phase

  // ---- stage 1: conv via table gather, result to LDS ----
  const float bx = conv_b[oc + 0], by = conv_b[oc + 1];
  const float bz = conv_b[oc + 2], bw = conv_b[oc + 3];
  for (int p = pr; p < TL; p += 2) {
    float ax = bx, ay = by, az = bz, aw = bw;
#pragma unroll
    for (int k = 0; k < KC; ++k) {
      int v = sid[p + k];
      if (v > 0) {   // v==0 row of T is exactly zero (emb padding_idx)
        const float4 tv = *(const float4*)(T + ((size_t)k * VOCAB + v) * DCH + oc);
        ax += tv.x; ay += tv.y; az += tv.z; aw += tv.w;
      }
    }
    if (t0 + p >= L) { ax = 0.f; ay = 0.f; az = 0.f; aw = 0.f; }  // zero-pad tail
    xt[p][oc + 0] = ax; xt[p][oc + 1] = ay;
    xt[p][oc + 2] = az; xt[p][oc + 3] = aw;
  }
  __syncthreads();

  // ---- stage 2: q[p] = dot(xt[p], score_w), wave32 shuffle reduction ----
  {
    const int wave = tid >> 5;
    const int lane = tid & 31;
    float swr[DCH / 32];
#pragma unroll
    for (int j = 0; j < DCH / 32; ++j) swr[j] = sw[lane + 32 * j];
    for (int p = wave; p < TL; p += 8) {
      float s = 0.f;
#pragma unroll
      for (int j = 0; j < DCH / 32; ++j) s += xt[p][lane + 32 * j] * swr[j];
#pragma unroll
      for (int off = 16; off > 0; off >>= 1) s += __shfl_xor(s, off, 32);
      if (lane == 0) q[p] = s;
    }
  }
  __syncthreads();

  // ---- stage 3: per-position softmax over m=1..4 (fold 1/m mean) ----
  if (tid < TL) {
    const int l = tid;
    const int j2 = (l >> 1) << 1;
    const int j3 = (l / 3) * 3;
    const int j4 = (l >> 2) << 2;
    float s1 = q[l];
    float s2 = 0.5f * (q[j2] + q[j2 + 1]);
    float s3 = (q[j3] + q[j3 + 1] + q[j3 + 2]) * (1.f / 3.f);
    float s4 = 0.25f * (q[j4] + q[j4 + 1] + q[j4 + 2] + q[j4 + 3]);
    float mx = fmaxf(fmaxf(s1, s2), fmaxf(s3, s4));
    float e1 = __expf(s1 - mx), e2 = __expf(s2 - mx);
    float e3 = __expf(s3 - mx), e4 = __expf(s4 - mx);
    float inv = 1.f / (e1 + e2 + e3 + e4);
    wc[l][0] = e1 * inv;
    wc[l][1] = e2 * inv * 0.5f;
    wc[l][2] = e3 * inv * (1.f / 3.f);
    wc[l][3] = e4 * inv * 0.25f;
  }
  __syncthreads();

  // ---- stage 4: latent mix + DS=2 mean pool, write out ----
  for (int l2 = pr; l2 < TL / 2; l2 += 2) {
    if (t0 + 2 * l2 >= L) break;
    float ax = 0.f, ay = 0.f, az = 0.f, aw = 0.f;
#pragma unroll
    for (int dl = 0; dl < 2; ++dl) {
      const int l  = 2 * l2 + dl;
      const int j2 = (l >> 1) << 1;
      const int j3 = (l / 3) * 3;
      const int j4 = (l >> 2) << 2;
      const float w1 = wc[l][0], w2 = wc[l][1], w3 = wc[l][2], w4 = wc[l][3];
      ax += w1 * xt[l][oc + 0];
      ay += w1 * xt[l][oc + 1];
      az += w1 * xt[l][oc + 2];
      aw += w1 * xt[l][oc + 3];
      ax += w2 * (xt[j2][oc + 0] + xt[j2 + 1][oc + 0]);
      ay += w2 * (xt[j2][oc + 1] + xt[j2 + 1][oc + 1]);
      az += w2 * (xt[j2][oc + 2] + xt[j2 + 1][oc + 2]);
      aw += w2 * (xt[j2][oc + 3] + xt[j2 + 1][oc + 3]);
      ax += w3 * (xt[j3][oc + 0] + xt[j3 + 1][oc + 0] + xt[j3 + 2][oc + 0]);
      ay += w3 * (xt[j3][oc + 1] + xt[j3 + 1][oc + 1] + xt[j3 + 2][oc + 1]);
      az += w3 * (xt[j3][oc + 2] + xt[j3 + 1][oc + 2] + xt[j3 + 2][oc + 2]);
      aw += w3 * (xt[j3][oc + 3] + xt[j3 + 1][oc + 3] + xt[j3 + 2][oc + 3]);
      ax += w4 * (xt[j4][oc + 0] + xt[j4 + 1][oc + 0] + xt[j4 + 2][oc + 0] + xt[j4 + 3][oc + 0]);
      ay += w4 * (xt[j4][oc + 1] + xt[j4 + 1][oc + 1] + xt[j4 + 2][oc + 1] + xt[j4 + 3][oc + 1]);
      az += w4 * (xt[j4][oc + 2] + xt[j4 + 1][oc + 2] + xt[j4 + 2][oc + 2] + xt[j4 + 3][oc + 2]);
      aw += w4 * (xt[j4][oc + 3] + xt[j4 + 1][oc + 3] + xt[j4 + 2][oc + 3] + xt[j4 + 3][oc + 3]);
    }
    float4 r;
    r.x = 0.5f * ax; r.y = 0.5f * ay; r.z = 0.5f * az; r.w = 0.5f * aw;
    const size_t off = ((size_t)b * (L >> 1) + ((t0 >> 1) + l2)) * DCH + oc;
    *(float4*)(out + off) = r;
  }
}

// -----------------------------------------------------------------------------
extern "C" void kernel_launch(void* const* d_in, const int* in_sizes, int n_in,
                              void* d_out, int out_size, void* d_ws, size_t ws_size,
                              hipStream_t stream) {
  const int*   ids     = (const int*)d_in[0];
  const float* emb     = (const float*)d_in[1];
  const float* conv_w  = (const float*)d_in[2];
  const float* conv_b  = (const float*)d_in[3];
  const float* score_w = (const float*)d_in[4];
  float* out = (float*)d_out;
  float* T   = (float*)d_ws;             // KC*VOCAB*DCH floats = 2.62 MB

  const int L = LLEN;
  const int B = in_sizes[0] / L;

  // Table precompute: 5 GEMMs of 256x512x512 on the f32 WMMA path.
  dim3 g1(DCH / 16, VOCAB / 16, KC);     // (32, 16, 5)
  gbst_table_wmma<<<g1, 32, 0, stream>>>(emb, conv_w, T);

  // Fused GBST: one block per 48-position tile per batch row.
  dim3 g2((L + TL - 1) / TL, B);         // (171, 8)
  gbst_main<<<g2, 256, 0, stream>>>(ids, T, conv_b, score_w, out, L);
}